// SoftRedProcessor_63350767616607
// MI455X (gfx1250) — compile-verified
//
#include <hip/hip_runtime.h>
#include <stdint.h>

#define VOCAB   128000
#define HALF_V  64000
#define TSEED   42u

typedef __attribute__((ext_vector_type(4))) float v4f;

struct U2 { uint32_t x, y; };

__device__ __forceinline__ uint32_t rotl32(uint32_t v, uint32_t r) {
    return __builtin_rotateleft32(v, r);   // lowers to v_alignbit_b32
}

// Threefry-2x32, 20 rounds (Random123 / JAX-compatible)
__device__ __forceinline__ U2 threefry2x32(uint32_t k0, uint32_t k1,
                                           uint32_t x0, uint32_t x1) {
    const uint32_t k2 = k0 ^ k1 ^ 0x1BD11BDAu;
    x0 += k0; x1 += k1;
#define TF_R(r) { x0 += x1; x1 = rotl32(x1, r); x1 ^= x0; }
    TF_R(13) TF_R(15) TF_R(26) TF_R(6)
    x0 += k1; x1 += k2 + 1u;
    TF_R(17) TF_R(29) TF_R(16) TF_R(24)
    x0 += k2; x1 += k0 + 2u;
    TF_R(13) TF_R(15) TF_R(26) TF_R(6)
    x0 += k0; x1 += k1 + 3u;
    TF_R(17) TF_R(29) TF_R(16) TF_R(24)
    x0 += k1; x1 += k2 + 4u;
    TF_R(13) TF_R(15) TF_R(26) TF_R(6)
    x0 += k2; x1 += k0 + 5u;
#undef TF_R
    U2 r; r.x = x0; r.y = x1; return r;
}

// Kernel 1: one thread per row -> derive the "lower bits" key k2 of jax.random.split
__global__ __launch_bounds__(256)
void rowkey_kernel(const int* __restrict__ ids, uint2* __restrict__ keys,
                   int batch, int seq) {
    int b = blockIdx.x * blockDim.x + threadIdx.x;
    if (b >= batch) return;
    const int* p = ids + (size_t)b * seq + (seq - 3);
    uint32_t s = ((uint32_t)p[0] + (uint32_t)p[1] + (uint32_t)p[2]) * TSEED;
    // fold_in(key(42), s) : threefry((0,42), (0, s))
    U2 rk = threefry2x32(0u, TSEED, 0u, s);
    // split(rk, 2): counts iota(4) -> pairs (0,2) and (1,3); k2 = (A.y, B.y)
    U2 A = threefry2x32(rk.x, rk.y, 0u, 2u);
    U2 B = threefry2x32(rk.x, rk.y, 1u, 3u);
    keys[b] = make_uint2(A.y, B.y);
}

// Kernel 2: grid = (ceil(64000/1024), batch); each thread: 4 consecutive i in [0,64000),
// one threefry per i covers vocab positions i and i+64000.
__global__ __launch_bounds__(256)
void bias_kernel(const float* __restrict__ scores,
                 const uint2* __restrict__ keys,
                 float* __restrict__ out) {
    const int   b  = blockIdx.y;
    const uint2 k  = keys[b];                        // uniform -> s_load_b64
    const int   i0 = (blockIdx.x * 256 + threadIdx.x) * 4;
    if (i0 >= HALF_V) return;                        // 64000 % 4 == 0: all-or-nothing

    const size_t row = (size_t)b * VOCAB;
    const v4f slo = __builtin_nontemporal_load((const v4f*)(scores + row + i0));
    const v4f shi = __builtin_nontemporal_load((const v4f*)(scores + row + i0 + HALF_V));

    // 4 independent hash chains -> ILP on the SIMD32
    U2 r0 = threefry2x32(k.x, k.y, (uint32_t)(i0 + 0), (uint32_t)(i0 + 0 + HALF_V));
    U2 r1 = threefry2x32(k.x, k.y, (uint32_t)(i0 + 1), (uint32_t)(i0 + 1 + HALF_V));
    U2 r2 = threefry2x32(k.x, k.y, (uint32_t)(i0 + 2), (uint32_t)(i0 + 2 + HALF_V));
    U2 r3 = threefry2x32(k.x, k.y, (uint32_t)(i0 + 3), (uint32_t)(i0 + 3 + HALF_V));

    // (bit & 1) << 30 == 0x40000000 == 2.0f as IEEE-754; bit-exact vs scores + 2*mask
    v4f olo, ohi;
    olo.x = slo.x + __uint_as_float((r0.x & 1u) << 30);
    olo.y = slo.y + __uint_as_float((r1.x & 1u) << 30);
    olo.z = slo.z + __uint_as_float((r2.x & 1u) << 30);
    olo.w = slo.w + __uint_as_float((r3.x & 1u) << 30);
    ohi.x = shi.x + __uint_as_float((r0.y & 1u) << 30);
    ohi.y = shi.y + __uint_as_float((r1.y & 1u) << 30);
    ohi.z = shi.z + __uint_as_float((r2.y & 1u) << 30);
    ohi.w = shi.w + __uint_as_float((r3.y & 1u) << 30);

    __builtin_nontemporal_store(olo, (v4f*)(out + row + i0));
    __builtin_nontemporal_store(ohi, (v4f*)(out + row + i0 + HALF_V));
}

extern "C" void kernel_launch(void* const* d_in, const int* in_sizes, int n_in,
                              void* d_out, int out_size, void* d_ws, size_t ws_size,
                              hipStream_t stream) {
    const int*   ids    = (const int*)d_in[0];     // (B, SEQ) int32
    const float* scores = (const float*)d_in[1];   // (B, VOCAB) f32
    float*       out    = (float*)d_out;           // (B, VOCAB) f32
    uint2*       keys   = (uint2*)d_ws;            // B * 8 bytes

    const int batch = in_sizes[1] / VOCAB;
    const int seq   = in_sizes[0] / batch;

    rowkey_kernel<<<(batch + 255) / 256, 256, 0, stream>>>(ids, keys, batch, seq);

    dim3 grid((HALF_V + 1023) / 1024, batch);
    bias_kernel<<<grid, 256, 0, stream>>>(scores, keys, out);
}